// DiffAttention_81063212745635
// MI455X (gfx1250) — compile-verified
//
#include <hip/hip_runtime.h>

// ---------------------------------------------------------------------------
// DiffAttention on MI455X (gfx1250, wave32, WMMA + async global->LDS DMA).
// Pipeline: 3x proj_gemm (Q,K,V) -> 2x flash attention (components 1,2)
//           -> combine (O1 - lam*O2) -> groupnorm stats -> normalize+transpose
//           -> proj_gemm (Wo).
// All matmuls use V_WMMA_F32_16X16X4_F32 (fp32 in, fp32 accum).
// LDS tiles staged with GLOBAL_LOAD_ASYNC_TO_LDS_B128 (ASYNCcnt) where the
// copy is layout-preserving; K tiles are transposed on the VALU path which
// now overlaps with the in-flight async DMA.
// Workspace: ~112 MB fp32 scratch in d_ws.
// ---------------------------------------------------------------------------

typedef float v2f __attribute__((ext_vector_type(2)));
typedef float v8f __attribute__((ext_vector_type(8)));

__device__ __forceinline__ v8f wmma_f32(v2f a, v2f b, v8f c) {
  // 8-arg form: (neg_a, A, neg_b, B, c_mod, C, reuse_a, reuse_b)
  return __builtin_amdgcn_wmma_f32_16x16x4_f32(false, a, false, b, (short)0, c,
                                               false, false);
}

// Async 16B global->LDS copy (per active lane). lds_byte_off is the
// workgroup-relative LDS address (low 32 bits of the flat shared pointer).
__device__ __forceinline__ void async_copy16(unsigned lds_byte_off,
                                             const void* gaddr) {
  asm volatile("global_load_async_to_lds_b128 %0, %1, off"
               :: "v"(lds_byte_off), "v"(gaddr)
               : "memory");
}

__device__ __forceinline__ void wait_async0() {
  asm volatile("s_wait_asynccnt 0x0" ::: "memory");
}

__device__ __forceinline__ unsigned lds_off(const void* p) {
  return (unsigned)(unsigned long long)p;  // flat addr low 32 bits == LDS offset
}

constexpr int Bsz  = 2;
constexpr int Seq  = 2048;
constexpr int Dh   = 128;   // head dim
constexpr int NH   = 8;
constexpr int HDim = NH * Dh;      // 1024
constexpr int QKW  = 2 * HDim;     // 2048 (row width of Q/K projections)

// ---------------------------------------------------------------------------
// Generic GEMM: C[M,N] = A[M,K] @ W[N,K]^T + bias[N].  M,N,K multiples of 64.
// Block: 128 threads (4 waves). Block tile 64x64, wave tile 32x32.
// ---------------------------------------------------------------------------
__global__ __launch_bounds__(128) void proj_gemm(const float* __restrict__ A,
                                                 const float* __restrict__ W,
                                                 const float* __restrict__ bias,
                                                 float* __restrict__ C,
                                                 int M, int N, int K) {
  // pad 68: rows stay 16B-aligned (68*4 = 272 = 17*16) and fragment reads
  // (row varies per lane, column fixed) land on distinct banks.
  __shared__ float As[64][68];
  __shared__ float Ws[64][68];

  const int tid  = threadIdx.x;
  const int wave = tid >> 5;
  const int lane = tid & 31;
  const int half = lane >> 4;     // 0: lanes 0-15, 1: lanes 16-31
  const int l16  = lane & 15;

  const int m0 = blockIdx.x * 64;
  const int n0 = blockIdx.y * 64;
  const int wm = (wave >> 1) * 32;  // wave tile row offset within block
  const int wn = (wave & 1) * 32;   // wave tile col offset

  v8f acc[2][2];
#pragma unroll
  for (int i = 0; i < 2; ++i)
#pragma unroll
    for (int j = 0; j < 2; ++j) {
      v8f z = {0.f, 0.f, 0.f, 0.f, 0.f, 0.f, 0.f, 0.f};
      acc[i][j] = z;
    }

  for (int k0 = 0; k0 < K; k0 += 64) {
    __syncthreads();
    // Async-DMA the 64x64 A and W tiles into LDS (16B per lane per issue).
#pragma unroll
    for (int it = 0; it < 8; ++it) {
      int idx4 = tid + it * 128;          // [0,1024): 16 float4 per row
      int r = idx4 >> 4;
      int c = (idx4 & 15) << 2;
      async_copy16(lds_off(&As[r][c]), A + (size_t)(m0 + r) * K + k0 + c);
      async_copy16(lds_off(&Ws[r][c]), W + (size_t)(n0 + r) * K + k0 + c);
    }
    // Warm L2 for the next K-chunk while this one computes.
    if (k0 + 64 < K) {
      __builtin_prefetch(A + (size_t)(m0 + (tid >> 1)) * K + k0 + 64 + (tid & 1) * 32, 0, 0);
      __builtin_prefetch(W + (size_t)(n0 + (tid >> 1)) * K + k0 + 64 + (tid & 1) * 32, 0, 0);
    }
    wait_async0();
    __syncthreads();

#pragma unroll 4
    for (int kk = 0; kk < 64; kk += 4) {
      v2f a[2], b[2];
#pragma unroll
      for (int mf = 0; mf < 2; ++mf) {
        a[mf].x = As[wm + mf * 16 + l16][kk + 2 * half];
        a[mf].y = As[wm + mf * 16 + l16][kk + 2 * half + 1];
      }
#pragma unroll
      for (int nf = 0; nf < 2; ++nf) {
        b[nf].x = Ws[wn + nf * 16 + l16][kk + 2 * half];
        b[nf].y = Ws[wn + nf * 16 + l16][kk + 2 * half + 1];
      }
#pragma unroll
      for (int mf = 0; mf < 2; ++mf)
#pragma unroll
        for (int nf = 0; nf < 2; ++nf)
          acc[mf][nf] = wmma_f32(a[mf], b[nf], acc[mf][nf]);
    }
  }

  // Epilogue: C/D layout: vgpr j -> row j (lanes 0-15) / j+8 (lanes 16-31)
#pragma unroll
  for (int mf = 0; mf < 2; ++mf)
#pragma unroll
    for (int nf = 0; nf < 2; ++nf) {
      int col = n0 + wn + nf * 16 + l16;
      float bval = bias[col];
#pragma unroll
      for (int j = 0; j < 8; ++j) {
        int row = m0 + wm + mf * 16 + j + 8 * half;
        C[(size_t)row * N + col] = acc[mf][nf][j] + bval;
      }
    }
}

// ---------------------------------------------------------------------------
// Flash attention, one component (q1k1 or q2k2), fp32 WMMA.
// Grid: (B*H, Seq/64). Block: 128 threads (4 waves), each wave owns 16 queries.
// Qf/Kf: flat (B,S,2HD); q[b,h] = contiguous (S x 256) block; compOff selects
// the first/second 128 columns. Vf: flat (B,S,HD); v[b,h] = (S x 128) block.
// Output written in "F" layout: Obuf[b][ s*HD + h*D + d ].
// ---------------------------------------------------------------------------
__global__ __launch_bounds__(128) void diff_flash(const float* __restrict__ Qf,
                                                  const float* __restrict__ Kf,
                                                  const float* __restrict__ Vf,
                                                  float* __restrict__ Obuf,
                                                  int compOff) {
  __shared__ float KT[128][64];   // K tile transposed: KT[d][key]; reused for P
  __shared__ float Vt[64][128];   // V tile row-major: Vt[key][d]

  const int bh   = blockIdx.x;
  const int b    = bh >> 3;
  const int h    = bh & 7;
  const int wave = threadIdx.x >> 5;
  const int lane = threadIdx.x & 31;
  const int half = lane >> 4;
  const int l16  = lane & 15;

  const float* Qb = Qf + (size_t)b * Seq * QKW + (size_t)h * Seq * 256 + compOff;
  const float* Kb = Kf + (size_t)b * Seq * QKW + (size_t)h * Seq * 256 + compOff;
  const float* Vb = Vf + (size_t)b * Seq * HDim + (size_t)h * Seq * Dh;

  const int q0 = blockIdx.y * 64 + wave * 16;  // first query row of this wave

  // Load this wave's 16 Q rows into registers in A-fragment lane order:
  // lanes 0-15 hold K = 0,1 (mod 4); lanes 16-31 hold K = 2,3 (mod 4).
  float qreg[64];
  {
    const float* qrow = Qb + (size_t)(q0 + l16) * 256 + 2 * half;
#pragma unroll
    for (int j = 0; j < 32; ++j) {
      qreg[2 * j]     = qrow[4 * j];
      qreg[2 * j + 1] = qrow[4 * j + 1];
    }
  }

  v8f o[8];
#pragma unroll
  for (int df = 0; df < 8; ++df) {
    v8f z = {0.f, 0.f, 0.f, 0.f, 0.f, 0.f, 0.f, 0.f};
    o[df] = z;
  }
  float mrun[8], lrun[8];
#pragma unroll
  for (int j = 0; j < 8; ++j) { mrun[j] = -1e30f; lrun[j] = 0.f; }

  for (int kt = 0; kt < Seq / 64; ++kt) {
    __syncthreads();  // everyone done reading KT (P) / Vt from previous tile

    const float* Ksrc = Kb + (size_t)kt * 64 * 256;
    const float* Vsrc = Vb + (size_t)kt * 64 * 128;

    // V tile is a layout-preserving copy -> async DMA straight into LDS.
#pragma unroll
    for (int it = 0; it < 16; ++it) {
      int idx4 = threadIdx.x + it * 128;   // [0,2048): row = idx4/32, col4 = idx4%32
      int r = idx4 >> 5;
      int c = (idx4 & 31) << 2;
      async_copy16(lds_off(&Vt[r][c]), Vsrc + (size_t)r * 128 + c);
    }
    // K tile needs a transpose -> VALU path, overlapping the V DMA.
#pragma unroll
    for (int it = 0; it < 16; ++it) {
      int idx4 = threadIdx.x + it * 128;
      int r = idx4 >> 5;
      int c = (idx4 & 31) << 2;
      float4 kv = *(const float4*)(Ksrc + (size_t)r * 256 + c);
      KT[c + 0][r] = kv.x; KT[c + 1][r] = kv.y;
      KT[c + 2][r] = kv.z; KT[c + 3][r] = kv.w;
    }
    // Warm L2 for the next tile while this one computes.
    if (kt + 1 < Seq / 64) {
      __builtin_prefetch(Ksrc + 64 * 256 + threadIdx.x * 128, 0, 0);
      __builtin_prefetch(Vsrc + 64 * 128 + threadIdx.x * 64, 0, 0);
    }
    wait_async0();
    __syncthreads();

    // Scores: S(16x64) = Q(16x128) @ Ktile^T(128x64), 4 n-fragments.
    v8f s[4];
#pragma unroll
    for (int nf = 0; nf < 4; ++nf) {
      v8f z = {0.f, 0.f, 0.f, 0.f, 0.f, 0.f, 0.f, 0.f};
      s[nf] = z;
    }
#pragma unroll 4
    for (int kk = 0; kk < 128; kk += 4) {
      v2f a;
      a.x = qreg[kk / 2];
      a.y = qreg[kk / 2 + 1];
#pragma unroll
      for (int nf = 0; nf < 4; ++nf) {
        v2f bf;
        bf.x = KT[kk + 2 * half][nf * 16 + l16];
        bf.y = KT[kk + 2 * half + 1][nf * 16 + l16];
        s[nf] = wmma_f32(a, bf, s[nf]);
      }
    }

    // Online softmax update. vgpr j of C-frags = row j + 8*half; columns span
    // the 16 lanes of each half, so row reductions are xor-shuffles over 1..8.
#pragma unroll
    for (int j = 0; j < 8; ++j) {
      float mt = -1e30f;
#pragma unroll
      for (int nf = 0; nf < 4; ++nf) mt = fmaxf(mt, s[nf][j]);
#pragma unroll
      for (int msk = 8; msk >= 1; msk >>= 1) mt = fmaxf(mt, __shfl_xor(mt, msk, 32));
      float mnew = fmaxf(mrun[j], mt);
      float r = __expf(mrun[j] - mnew);
      mrun[j] = mnew;
      float rowsum = 0.f;
#pragma unroll
      for (int nf = 0; nf < 4; ++nf) {
        float p = __expf(s[nf][j] - mnew);
        s[nf][j] = p;
        rowsum += p;
      }
#pragma unroll
      for (int msk = 8; msk >= 1; msk >>= 1) rowsum += __shfl_xor(rowsum, msk, 32);
      lrun[j] = lrun[j] * r + rowsum;
#pragma unroll
      for (int df = 0; df < 8; ++df) o[df][j] *= r;
    }

    // Bounce P through LDS (KT region is dead now) to get A-fragment layout.
    __syncthreads();
    float* Pw = &KT[0][0] + wave * 1024;  // per-wave 16x64, stride 64
#pragma unroll
    for (int j = 0; j < 8; ++j)
#pragma unroll
      for (int nf = 0; nf < 4; ++nf)
        Pw[(j + 8 * half) * 64 + nf * 16 + l16] = s[nf][j];
    __syncthreads();

    // O(16x128) += P(16x64) @ Vtile(64x128)
#pragma unroll 2
    for (int kk = 0; kk < 64; kk += 4) {
      v2f a;
      a.x = Pw[l16 * 64 + kk + 2 * half];
      a.y = Pw[l16 * 64 + kk + 2 * half + 1];
#pragma unroll
      for (int df = 0; df < 8; ++df) {
        v2f bf;
        bf.x = Vt[kk + 2 * half][df * 16 + l16];
        bf.y = Vt[kk + 2 * half + 1][df * 16 + l16];
        o[df] = wmma_f32(a, bf, o[df]);
      }
    }
  }

  // Finalize (divide by l) and store in F layout: Obuf[b][s*HD + h*D + d].
  float* Ob = Obuf + (size_t)b * Seq * HDim;
#pragma unroll
  for (int j = 0; j < 8; ++j) {
    float inv = 1.f / lrun[j];
    int srow = q0 + j + 8 * half;
#pragma unroll
    for (int df = 0; df < 8; ++df)
      Ob[(size_t)srow * HDim + h * Dh + df * 16 + l16] = o[df][j] * inv;
  }
}

// ---------------------------------------------------------------------------
// F = O1 - lam * O2 (in place into O1)
// ---------------------------------------------------------------------------
__global__ void combine_kernel(float* __restrict__ O1, const float* __restrict__ O2,
                               const float* __restrict__ lam, int n) {
  int i = blockIdx.x * blockDim.x + threadIdx.x;
  if (i < n) O1[i] = O1[i] - (*lam) * O2[i];
}

// ---------------------------------------------------------------------------
// Group-norm stats: 16 blocks, each reduces a contiguous chunk of S*D = 262144
// floats of F (chunk index = b*8 + h2). Writes (mean, rstd) pairs.
// ---------------------------------------------------------------------------
__global__ __launch_bounds__(256) void gn_stats(const float* __restrict__ F,
                                                float* __restrict__ stats) {
  __shared__ float ssum[256];
  __shared__ float ssq[256];
  const float* p = F + (size_t)blockIdx.x * (Seq * Dh);
  float s = 0.f, q = 0.f;
  for (int i = threadIdx.x; i < Seq * Dh; i += 256) {
    float v = p[i];
    s += v;
    q += v * v;
  }
  ssum[threadIdx.x] = s;
  ssq[threadIdx.x] = q;
  __syncthreads();
  for (int off = 128; off > 0; off >>= 1) {
    if (threadIdx.x < off) {
      ssum[threadIdx.x] += ssum[threadIdx.x + off];
      ssq[threadIdx.x]  += ssq[threadIdx.x + off];
    }
    __syncthreads();
  }
  if (threadIdx.x == 0) {
    float inv_n = 1.f / (float)(Seq * Dh);
    float mean = ssum[0] * inv_n;
    float var  = ssq[0] * inv_n - mean * mean;
    stats[blockIdx.x * 2]     = mean;
    stats[blockIdx.x * 2 + 1] = rsqrtf(var + 1e-5f);
  }
}

// ---------------------------------------------------------------------------
// Normalize + final transpose: Z[b][s*HD + h*D + d] =
//   ((F[b][h*S*D + s*D + d] - mean[b,h]) * rstd[b,h] * gnw[h] + gnb[h]) * (1-lam)
// ---------------------------------------------------------------------------
__global__ __launch_bounds__(256) void norm_transpose(const float* __restrict__ F,
                                                      const float* __restrict__ stats,
                                                      const float* __restrict__ gnw,
                                                      const float* __restrict__ gnb,
                                                      const float* __restrict__ lam,
                                                      float* __restrict__ Z) {
  size_t e = ((size_t)blockIdx.x * blockDim.x + threadIdx.x) * 4;  // 4 floats per thread
  int b  = (int)(e / ((size_t)Seq * HDim));
  size_t eb = e % ((size_t)Seq * HDim);
  int s3 = (int)(eb / HDim);
  int c  = (int)(eb % HDim);
  int h3 = c / Dh;
  int d3 = c % Dh;
  float4 x = *(const float4*)(F + (size_t)b * Seq * HDim + (size_t)h3 * Seq * Dh +
                              (size_t)s3 * Dh + d3);
  float mean = stats[(b * 8 + h3) * 2];
  float rstd = stats[(b * 8 + h3) * 2 + 1];
  float oml  = 1.f - (*lam);
  float scale = rstd * gnw[h3] * oml;
  float shift = (gnb[h3] - mean * rstd * gnw[h3]) * oml;
  float4 y;
  y.x = x.x * scale + shift;
  y.y = x.y * scale + shift;
  y.z = x.z * scale + shift;
  y.w = x.w * scale + shift;
  *(float4*)(Z + e) = y;
}

// ---------------------------------------------------------------------------
extern "C" void kernel_launch(void* const* d_in, const int* in_sizes, int n_in,
                              void* d_out, int out_size, void* d_ws, size_t ws_size,
                              hipStream_t stream) {
  const float* query = (const float*)d_in[0];
  const float* Wq    = (const float*)d_in[1];
  const float* bq    = (const float*)d_in[2];
  const float* Wk    = (const float*)d_in[3];
  const float* bk    = (const float*)d_in[4];
  const float* Wv    = (const float*)d_in[5];
  const float* bv    = (const float*)d_in[6];
  const float* Wo    = (const float*)d_in[7];
  const float* bo    = (const float*)d_in[8];
  const float* gnw   = (const float*)d_in[9];
  const float* gnb   = (const float*)d_in[10];
  const float* lam   = (const float*)d_in[11];

  // Workspace layout (floats):
  float* Qf = (float*)d_ws;                         // B*S*2HD = 8,388,608
  float* Kf = Qf + (size_t)Bsz * Seq * QKW;         // 8,388,608
  float* Vf = Kf + (size_t)Bsz * Seq * QKW;         // B*S*HD  = 4,194,304
  float* O1 = Vf + (size_t)Bsz * Seq * HDim;        // 4,194,304
  float* O2 = O1 + (size_t)Bsz * Seq * HDim;        // 4,194,304
  float* stats = O2 + (size_t)Bsz * Seq * HDim;     // 32
  float* Z  = Qf;  // reuse Qf region after attention is done

  const int M = Bsz * Seq;  // 4096
  dim3 blk(128);

  proj_gemm<<<dim3(M / 64, QKW / 64), blk, 0, stream>>>(query, Wq, bq, Qf, M, QKW, Dh);
  proj_gemm<<<dim3(M / 64, QKW / 64), blk, 0, stream>>>(query, Wk, bk, Kf, M, QKW, Dh);
  proj_gemm<<<dim3(M / 64, HDim / 64), blk, 0, stream>>>(query, Wv, bv, Vf, M, HDim, Dh);

  diff_flash<<<dim3(Bsz * NH, Seq / 64), blk, 0, stream>>>(Qf, Kf, Vf, O1, 0);
  diff_flash<<<dim3(Bsz * NH, Seq / 64), blk, 0, stream>>>(Qf, Kf, Vf, O2, Dh);

  int nEl = Bsz * Seq * HDim;  // 4,194,304
  combine_kernel<<<dim3(nEl / 256), dim3(256), 0, stream>>>(O1, O2, lam, nEl);
  gn_stats<<<dim3(Bsz * NH), dim3(256), 0, stream>>>(O1, stats);
  norm_transpose<<<dim3(nEl / 4 / 256), dim3(256), 0, stream>>>(O1, stats, gnw, gnb, lam, Z);

  proj_gemm<<<dim3(M / 64, Dh / 64), blk, 0, stream>>>(Z, Wo, bo, (float*)d_out, M, Dh, HDim);
}